// Decoder_67705864454693
// MI455X (gfx1250) — compile-verified
//
#include <hip/hip_runtime.h>
#include <math.h>

typedef __attribute__((ext_vector_type(2))) float v2f;
typedef __attribute__((ext_vector_type(8))) float v8f;

#define HSTRIDE 130   // 16-row LDS activation stride; 130 % 64 != 0 -> no bank conflicts

// ---------------------------------------------------------------------------
// Kernel 1: four MLP heads via V_WMMA_F32_16X16X4_F32.
// grid = (64 M-tiles, 4 heads), block = 256 (8 waves; wave w owns N cols 16w..16w+15)
// ---------------------------------------------------------------------------
__global__ __launch_bounds__(256) void decoder_heads_wmma(
    const float* __restrict__ x,      // (1024,128) = (B*E, D)
    const float* __restrict__ Ws,     // (4,3,128,128)  [o,d] row-major
    const float* __restrict__ bs,     // (4,3,128)
    const float* __restrict__ Wo,     // (4,1,128)
    const float* __restrict__ bo,     // (4,1)
    float* __restrict__ params)       // (4,1024): loc | var | freq | amp
{
    __shared__ float hbuf[2][16 * HSTRIDE];
    __shared__ float red[256];

    const int tile = blockIdx.x;      // 16-row strip of the 1024-row activation
    const int head = blockIdx.y;
    const int tid  = threadIdx.x;
    const int lane = tid & 31;
    const int wv   = tid >> 5;
    const int half = lane >> 4;       // 0: K={0,1}, 1: K={2,3} within a 4-chunk
    const int l15  = lane & 15;
    const int klo  = half << 1;
    const int nb   = wv << 4;

    // stage the 16x128 strip of x into LDS
    for (int i = tid; i < 16 * 128; i += 256) {
        int m = i >> 7, k = i & 127;
        hbuf[0][m * HSTRIDE + k] = x[(tile * 16 + m) * 128 + k];
    }
    __syncthreads();

    int cur = 0;
    for (int layer = 0; layer < 3; ++layer) {
        const float* W    = Ws + (((head * 3) + layer) << 14);   // +layer*128*128
        const float* bias = bs + ((head * 3) + layer) * 128;
        const float* arow = &hbuf[cur][l15 * HSTRIDE + klo];     // A: h[m][k]
        const float* brow = &W[(nb + l15) * 128 + klo];          // B: W[n][k] (h*W^T)
        v8f c = {};
        #pragma unroll
        for (int kb = 0; kb < 128; kb += 4) {
            v2f a, b;
            a.x = arow[kb];
            a.y = arow[kb + 1];
            b.x = brow[kb];
            b.y = brow[kb + 1];
            // 8 args: (neg_a, A, neg_b, B, c_mod, C, reuse_a, reuse_b)
            c = __builtin_amdgcn_wmma_f32_16x16x4_f32(
                    false, a, false, b, (short)0, c, false, false);
        }
        // bias + leaky_relu(0.2), write 16x16 D tile into the other buffer
        float  bv   = bias[nb + l15];
        float* orow = &hbuf[cur ^ 1][(half * 8) * HSTRIDE + nb + l15];
        #pragma unroll
        for (int r = 0; r < 8; ++r) {
            float v = c[r] + bv;
            v = v > 0.0f ? v : 0.2f * v;
            orow[r * HSTRIDE] = v;
        }
        cur ^= 1;
        __syncthreads();
    }

    // final Linear(128 -> 1): 16 threads per row, 8 k's per thread, LDS reduce
    {
        const float* wo = Wo + head * 128;
        int r = tid >> 4, s = tid & 15;
        const float* hr = &hbuf[cur][r * HSTRIDE + s * 8];
        float p = 0.0f;
        #pragma unroll
        for (int j = 0; j < 8; ++j) p += hr[j] * wo[s * 8 + j];
        red[tid] = p;
        __syncthreads();
        if (tid < 16) {
            float v = bo[head];
            #pragma unroll
            for (int s2 = 0; s2 < 16; ++s2) v += red[tid * 16 + s2];
            float sg = 1.0f / (1.0f + __expf(-v));
            float outv;
            if (head == 0)      outv = sg;            // loc
            else if (head == 1) outv = 1e-8f + sg;    // var (Normal scale)
            else if (head == 2) outv = sg * sg;       // freq
            else                outv = sg;            // amp
            params[head * 1024 + tile * 16 + tid] = outv;
        }
    }
}

// ---------------------------------------------------------------------------
// Kernel 2: per-event derived constants for the synthesis inner loop.
//   q = { freq*pi, amp/(var*sqrt(2pi)), 1/var, loc/var }
// ---------------------------------------------------------------------------
__global__ __launch_bounds__(256) void decoder_derive(
    const float* __restrict__ params,   // (4,1024)
    float4* __restrict__ ev)            // (1024,)
{
    int i = blockIdx.x * 256 + threadIdx.x;   // 0..1023
    float loc  = params[i];
    float var  = params[1024 + i];
    float fr   = params[2048 + i];
    float amp  = params[3072 + i];
    float invv = 1.0f / var;
    float4 q;
    q.x = fr * 3.14159265358979323846f;
    q.y = amp * invv * 0.39894228040143268f;   // 1/sqrt(2*pi)
    q.z = invv;
    q.w = loc * invv;
    ev[i] = q;
}

// ---------------------------------------------------------------------------
// Kernel 3: synthesis. out[b,n] = sum_e coef_e * sin(fpi_e*(n+1)) * exp(-t^2/2)
// grid = 8*128 blocks of 256 threads (one thread per output sample).
// ---------------------------------------------------------------------------
__global__ __launch_bounds__(256) void decoder_synth(
    const float4* __restrict__ ev,    // (1024,)
    float* __restrict__ out)          // (8, 32768), unnormalized
{
    __shared__ float4 sev[128];
    const int b     = blockIdx.x >> 7;
    const int chunk = blockIdx.x & 127;
    const int tid   = threadIdx.x;
    if (tid < 128) sev[tid] = ev[b * 128 + tid];
    __syncthreads();

    const int   n    = chunk * 256 + tid;               // 0..32767
    const float ramp = (float)(n + 1);
    const float rng  = (float)n * (1.0f / 32767.0f);    // linspace(0,1,N)

    const float INV_2PI  = 0.15915494309189535f;
    const float TWOPI_HI = 6.28318548202514648f;        // float(2*pi)
    const float TWOPI_LO = -1.74845553e-7f;             // 2*pi - TWOPI_HI

    float acc = 0.0f;
    #pragma unroll 4
    for (int e = 0; e < 128; ++e) {
        float4 q = sev[e];
        float phase = q.x * ramp;
        // Cody-Waite reduction so v_sin_f32 stays in its valid range
        float kk = rintf(phase * INV_2PI);
        float rr = fmaf(kk, -TWOPI_HI, phase);
        rr       = fmaf(kk, -TWOPI_LO, rr);
        float s  = __sinf(rr);
        float t  = fmaf(rng, q.z, -q.w);                // (rng - loc)/var
        float g  = __expf(-0.5f * t * t);
        acc = fmaf(q.y * s, g, acc);
    }
    out[b * 32768 + n] = acc;
}

// ---------------------------------------------------------------------------
// Kernel 4: in-place max-abs normalization per batch item.
// ---------------------------------------------------------------------------
__global__ __launch_bounds__(256) void decoder_maxnorm(float* __restrict__ out)
{
    __shared__ float red[256];
    const int b = blockIdx.x;
    float* o = out + b * 32768;
    float m = 0.0f;
    for (int i = threadIdx.x; i < 32768; i += 256)
        m = fmaxf(m, fabsf(o[i]));
    red[threadIdx.x] = m;
    __syncthreads();
    for (int s = 128; s > 0; s >>= 1) {
        if (threadIdx.x < s)
            red[threadIdx.x] = fmaxf(red[threadIdx.x], red[threadIdx.x + s]);
        __syncthreads();
    }
    float inv = 1.0f / (red[0] + 1e-12f);
    for (int i = threadIdx.x; i < 32768; i += 256)
        o[i] *= inv;
}

// ---------------------------------------------------------------------------
extern "C" void kernel_launch(void* const* d_in, const int* in_sizes, int n_in,
                              void* d_out, int out_size, void* d_ws, size_t ws_size,
                              hipStream_t stream)
{
    (void)in_sizes; (void)n_in; (void)out_size; (void)ws_size;

    const float* x  = (const float*)d_in[0];   // (8,128,128)
    const float* Ws = (const float*)d_in[1];   // (4,3,128,128)
    const float* bs = (const float*)d_in[2];   // (4,3,128)
    const float* Wo = (const float*)d_in[3];   // (4,1,128)
    const float* bo = (const float*)d_in[4];   // (4,1)
    float* out = (float*)d_out;                // (8,1,32768)

    float*  params = (float*)d_ws;                                  // 4*1024 f32
    float4* ev     = (float4*)((char*)d_ws + 4096 * sizeof(float)); // 1024 float4

    decoder_heads_wmma<<<dim3(64, 4), 256, 0, stream>>>(x, Ws, bs, Wo, bo, params);
    decoder_derive<<<4, 256, 0, stream>>>(params, ev);
    decoder_synth<<<8 * 128, 256, 0, stream>>>(ev, out);
    decoder_maxnorm<<<8, 256, 0, stream>>>(out);
}